// SSKernelNPLR_28853590294573
// MI455X (gfx1250) — compile-verified
//
#include <hip/hip_runtime.h>

// SSKernelNPLR on gfx1250:
//   Kernel 1: Cauchy sum + rank-1 Woodbury -> interleaved real A-matrix (H x 4100) in d_ws
//   Kernel 2: irfft realized as f32 GEMM via V_WMMA_F32_16X16X4_F32 with 2x2 register
//             blocking (4 accumulators / wave); twiddles generated in-register with
//             exact power-of-two modular reduction (L = 2^12).

#define HH   512
#define NN   32
#define LL   4096
#define LF   2049          // LL/2 + 1
#define KPAD 4100          // 2*LF rounded up to a multiple of 4 (WMMA K step)

typedef float v2f __attribute__((ext_vector_type(2)));
typedef float v8f __attribute__((ext_vector_type(8)));

struct cf { float x, y; };
__device__ __forceinline__ cf cmul(cf a, cf b) { return {a.x*b.x - a.y*b.y, a.x*b.y + a.y*b.x}; }
__device__ __forceinline__ cf cinv(cf a) {
    float r = 1.0f / (a.x*a.x + a.y*a.y);
    return {a.x * r, -a.y * r};
}
__device__ __forceinline__ cf cconj(cf a) { return {a.x, -a.y}; }
__device__ __forceinline__ cf cscale(cf a, float s) { return {a.x*s, a.y*s}; }
__device__ __forceinline__ cf cmac(cf acc, cf a, cf b) {   // acc += a*b
    acc.x = fmaf(a.x, b.x, fmaf(-a.y, b.y, acc.x));
    acc.y = fmaf(a.x, b.y, fmaf( a.y, b.x, acc.y));
    return acc;
}

// ---------------------------------------------------------------------------
// Kernel 1: per-head Cauchy kernel + Woodbury, writes weighted interleaved
// A-matrix Af[h][2l] = w_l*Re(k_f), Af[h][2l+1] = -w_l*Im(k_f).
// ---------------------------------------------------------------------------
__global__ __launch_bounds__(256)
void ssk_cauchy_kernel(const float* __restrict__ log_dt,
                       const float* __restrict__ Bm,
                       const float* __restrict__ Pm,
                       const float* __restrict__ Cm,
                       const float* __restrict__ lwr,
                       const float* __restrict__ wim,
                       float* __restrict__ Af)
{
    __shared__ float s_wd[2][NN];       // wd = w*dt (re, im)
    __shared__ float s_v[4][2][NN];     // v[a*2+b] (re, im)

    const int h   = blockIdx.x;
    const int tid = threadIdx.x;
    const float dt = expf(log_dt[h]);

    if (tid < NN) {
        const int n    = tid;
        const int base = h * NN + n;
        const float wr = -expf(lwr[base]);
        const float wi = wim[base];
        s_wd[0][n] = wr * dt;
        s_wd[1][n] = wi * dt;
        cf Bc = {Bm[2*base], Bm[2*base + 1]};
        cf Pc = {Pm[2*base], Pm[2*base + 1]};
        cf Cc = {Cm[2*base], Cm[2*base + 1]};
        cf Qc = cconj(Pc);
        cf v00 = cmul(Bc, Cc), v01 = cmul(Bc, Qc);
        cf v10 = cmul(Pc, Cc), v11 = cmul(Pc, Qc);
        s_v[0][0][n] = v00.x; s_v[0][1][n] = v00.y;
        s_v[1][0][n] = v01.x; s_v[1][1][n] = v01.y;
        s_v[2][0][n] = v10.x; s_v[2][1][n] = v10.y;
        s_v[3][0][n] = v11.x; s_v[3][1][n] = v11.y;
    }
    __syncthreads();

    for (int l = tid; l < LF; l += 256) {
        // omega = exp(-2*pi*i*l/L); z = 2*(1-omega)/(1+omega)
        float so, co;
        sincosf(-6.28318530717958647f * (float)l / (float)LL, &so, &co);
        cf one_p = {1.0f + co, so};
        cf inv1p = cinv(one_p);
        cf z     = cscale(cmul({1.0f - co, -so}, inv1p), 2.0f);

        cf r00 = {0,0}, r01 = {0,0}, r10 = {0,0}, r11 = {0,0};
        #pragma unroll 4
        for (int n = 0; n < NN; ++n) {
            const float wdr = s_wd[0][n], wdi = s_wd[1][n];
            cf den  = cinv({z.x - wdr, z.y - wdi});   // 1/(z - wd)
            cf denc = cinv({z.x - wdr, z.y + wdi});   // 1/(z - conj(wd))
            cf v00 = {s_v[0][0][n], s_v[0][1][n]};
            cf v01 = {s_v[1][0][n], s_v[1][1][n]};
            cf v10 = {s_v[2][0][n], s_v[2][1][n]};
            cf v11 = {s_v[3][0][n], s_v[3][1][n]};
            r00 = cmac(cmac(r00, v00, den), cconj(v00), denc);
            r01 = cmac(cmac(r01, v01, den), cconj(v01), denc);
            r10 = cmac(cmac(r10, v10, den), cconj(v10), denc);
            r11 = cmac(cmac(r11, v11, den), cconj(v11), denc);
        }
        r00 = cscale(r00, dt); r01 = cscale(r01, dt);
        r10 = cscale(r10, dt); r11 = cscale(r11, dt);

        // Woodbury: kf = (r00 - r01*r10/(1+r11)) * 2/(1+omega)
        cf r11p = {1.0f + r11.x, r11.y};
        cf num  = cmul(r01, r10);
        cf kf   = cmul(num, cinv(r11p));
        kf.x = r00.x - kf.x; kf.y = r00.y - kf.y;
        kf = cmul(kf, cscale(inv1p, 2.0f));

        // irfft weights folded in; sin row carries the minus sign
        const float wl = (l == 0 || l == LL/2) ? (1.0f / (float)LL) : (2.0f / (float)LL);
        Af[h * KPAD + 2*l    ] =  wl * kf.x;
        Af[h * KPAD + 2*l + 1] = -wl * kf.y;
    }
    if (tid < 2) Af[h * KPAD + 2*LF + tid] = 0.0f;   // K padding
}

// ---------------------------------------------------------------------------
// Kernel 2: k[h,t] = sum_K Af[h,K] * T[K,t] with T[2l,t]=cos(2*pi*l*t/L),
// T[2l+1,t]=sin(2*pi*l*t/L), via V_WMMA_F32_16X16X4_F32.
// Each wave owns a 32(h) x 32(t) macro-tile: 4 accumulators, so each A
// fragment and each twiddle fragment feeds TWO WMMAs (2x2 blocking).
// ---------------------------------------------------------------------------
__global__ __launch_bounds__(256)
void ssk_idft_wmma_kernel(const float* __restrict__ Af,
                          float* __restrict__ out)
{
    const int wave = (blockIdx.x * blockDim.x + threadIdx.x) >> 5; // 0..2047
    const int lane = threadIdx.x & 31;
    const int hb = wave >> 7;           // 0..15  (32-row h macro-tile)
    const int tb = wave & 127;          // 0..127 (32-col t macro-tile)
    const int h0 = hb << 5, t0 = tb << 5;
    const int mlo = lane & 15;
    const int hi  = (lane >> 4) & 1;    // K split within the fragment

    // A fragments: rows h0+m (tile 0) and h0+16+m (tile 1); 8B-aligned pairs.
    const float* Arow0 = Af + (size_t)(h0 + mlo) * KPAD + 2 * hi;
    const float* Arow1 = Arow0 + (size_t)16 * KPAD;
    const int ta = t0 + mlo;            // B columns: ta (tile 0), ta+16 (tile 1)
    const int tbcol = ta + 16;
    const float step = 6.28318530717958647f / (float)LL;

    v8f acc00 = {}, acc01 = {}, acc10 = {}, acc11 = {};
    for (int kk = 0; kk < KPAD / 4; ++kk) {
        // A fragments: VGPR0 = A[m, 4kk+2hi], VGPR1 = A[m, 4kk+2hi+1]
        v2f a0 = *(const v2f*)(Arow0 + 4 * kk);
        v2f a1 = *(const v2f*)(Arow1 + 4 * kk);

        // B fragments: row K=4kk+2hi -> cos, row +1 -> sin; exact mod-L index
        const int l0 = 2 * kk + hi;
        const int p0 = (l0 * ta)    & (LL - 1);
        const int p1 = (l0 * tbcol) & (LL - 1);
        float s0, c0, s1, c1;
        __sincosf((float)p0 * step, &s0, &c0);
        __sincosf((float)p1 * step, &s1, &c1);
        v2f b0; b0.x = c0; b0.y = s0;
        v2f b1; b1.x = c1; b1.y = s1;

        acc00 = __builtin_amdgcn_wmma_f32_16x16x4_f32(false, a0, false, b0,
                                                      (short)0, acc00, false, false);
        acc01 = __builtin_amdgcn_wmma_f32_16x16x4_f32(false, a0, false, b1,
                                                      (short)0, acc01, false, false);
        acc10 = __builtin_amdgcn_wmma_f32_16x16x4_f32(false, a1, false, b0,
                                                      (short)0, acc10, false, false);
        acc11 = __builtin_amdgcn_wmma_f32_16x16x4_f32(false, a1, false, b1,
                                                      (short)0, acc11, false, false);
    }

    // D layout: VGPR i holds M = i + 8*hi, N = lane&15
    {
        float* o00 = out + (size_t)(h0 + 8 * hi) * LL + t0 + mlo;
        float* o01 = o00 + 16;                       // +16 columns
        float* o10 = o00 + (size_t)16 * LL;          // +16 rows
        float* o11 = o10 + 16;
        #pragma unroll
        for (int i = 0; i < 8; ++i) {
            o00[(size_t)i * LL] = acc00[i];
            o01[(size_t)i * LL] = acc01[i];
            o10[(size_t)i * LL] = acc10[i];
            o11[(size_t)i * LL] = acc11[i];
        }
    }
}

// ---------------------------------------------------------------------------
extern "C" void kernel_launch(void* const* d_in, const int* in_sizes, int n_in,
                              void* d_out, int out_size, void* d_ws, size_t ws_size,
                              hipStream_t stream) {
    (void)in_sizes; (void)n_in; (void)out_size; (void)ws_size;
    const float* log_dt = (const float*)d_in[0];
    const float* Bm     = (const float*)d_in[1];
    const float* Pm     = (const float*)d_in[2];
    const float* Cm     = (const float*)d_in[3];
    const float* lwr    = (const float*)d_in[4];
    const float* wim    = (const float*)d_in[5];
    float* Af  = (float*)d_ws;          // H * KPAD floats = 8.4 MB scratch
    float* out = (float*)d_out;         // (1, H, L) = 8 MB f32

    ssk_cauchy_kernel<<<HH, 256, 0, stream>>>(log_dt, Bm, Pm, Cm, lwr, wim, Af);

    // 2048 waves, one 32x32 macro-tile each; 8 waves per 256-thread block
    ssk_idft_wmma_kernel<<<256, 256, 0, stream>>>(Af, out);
}